// BlackBoxV2_14877766713678
// MI455X (gfx1250) — compile-verified
//
#include <hip/hip_runtime.h>
#include <hip/hip_bf16.h>

typedef __attribute__((ext_vector_type(2))) float v2f;
typedef __attribute__((ext_vector_type(8))) float v8f;

#define BSEQ 4
#define NTOK 2048
#define DD   128
#define VV   32000

// ---------------------------------------------------------------------------
// Kernel 1: embedding gather + sequential gated recurrence.
// One workgroup per batch sequence (independent). 512 threads:
// t -> (d = t&127, q = t>>7); q splits each K-dot into quarters of 32.
// All weights this thread ever needs are loop-invariant -> 96 VGPRs:
//   wreg[j] = W[d, q*32+j]
//   g1[j]   = gate_w[d, q*32+j]          (state half)
//   g2[j]   = gate_w[d, 128 + q*32+j]    (new_state half)
// Inner loop: ds_load_b128 state/new chunks -> pure register FMAs ->
// 4-way LDS partial reduction. No weight memory traffic in the loop.
// ---------------------------------------------------------------------------
__launch_bounds__(512, 1)
__global__ void rnn_kernel(const int*   __restrict__ input_ids,
                           const float* __restrict__ embed_table,
                           const float* __restrict__ W,       // [128][128]
                           const float* __restrict__ gate_w,  // [128][256]
                           const float* __restrict__ gate_b,
                           const int*   __restrict__ n_inner_p,
                           float*       __restrict__ outs)    // [B,N,D]
{
    __shared__ float sPart[4 * DD];
    __shared__ float sState[DD];
    __shared__ float sNew[DD];
    __shared__ float sTok[DD];

    const int t  = threadIdx.x;
    const int d  = t & (DD - 1);
    const int q  = t >> 7;                 // 0..3
    const int b  = blockIdx.x;
    const int n_inner = n_inner_p[0];
    const int k0 = q * 32;

    // ---- hoist loop-invariant weights into registers (96 VGPRs) ----
    float wreg[32], g1[32], g2[32];
    #pragma unroll
    for (int j = 0; j < 32; ++j) {
        wreg[j] = W[d * DD + k0 + j];
        g1[j]   = gate_w[d * (2 * DD) + k0 + j];
        g2[j]   = gate_w[d * (2 * DD) + DD + k0 + j];
    }
    const float gb = gate_b[d];

    if (t < DD) sState[t] = 0.0f;
    __syncthreads();

    const float4* sSt4 = (const float4*)(sState + k0);
    const float4* sNw4 = (const float4*)(sNew + k0);

    for (int n = 0; n < NTOK; ++n) {
        if (q == 0) {
            int id = input_ids[b * NTOK + n];
            sTok[d] = embed_table[(size_t)id * DD + d];
        }
        __syncthreads();                          // sTok visible; prev reads done

        for (int it = 0; it < n_inner; ++it) {
            // ---- pull this quarter's state chunk into registers (8x b128) ----
            float s[32];
            #pragma unroll
            for (int j4 = 0; j4 < 8; ++j4) {
                float4 v = sSt4[j4];
                s[j4 * 4 + 0] = v.x; s[j4 * 4 + 1] = v.y;
                s[j4 * 4 + 2] = v.z; s[j4 * 4 + 3] = v.w;
            }

            // ---- new_state partial ----
            float p = 0.0f;
            #pragma unroll
            for (int j = 0; j < 32; ++j) p = fmaf(s[j], wreg[j], p);
            sPart[q * DD + d] = p;
            __syncthreads();                      // A

            float full = sTok[d] + sPart[d] + sPart[DD + d]
                       + sPart[2 * DD + d] + sPart[3 * DD + d];
            // exact GELU: 0.5*x*(1+erf(x/sqrt(2)))
            float ns = 0.5f * full * (1.0f + erff(full * 0.70710678118654752f));
            if (q == 0) sNew[d] = ns;
            __syncthreads();                      // B (sNew visible, sPart reusable)

            // ---- gate partial: state half + new_state half ----
            float nw[32];
            #pragma unroll
            for (int j4 = 0; j4 < 8; ++j4) {
                float4 v = sNw4[j4];
                nw[j4 * 4 + 0] = v.x; nw[j4 * 4 + 1] = v.y;
                nw[j4 * 4 + 2] = v.z; nw[j4 * 4 + 3] = v.w;
            }
            float pg = 0.0f;
            #pragma unroll
            for (int j = 0; j < 32; ++j) {
                pg = fmaf(s[j],  g1[j], pg);
                pg = fmaf(nw[j], g2[j], pg);
            }
            sPart[q * DD + d] = pg;
            __syncthreads();                      // C (all sState reads done)

            if (q == 0) {
                float gs = gb + sPart[d] + sPart[DD + d]
                         + sPart[2 * DD + d] + sPart[3 * DD + d];
                float g  = 1.0f / (1.0f + expf(-gs));
                sState[d] = g * ns + (1.0f - g) * sState[d];
            }
            __syncthreads();                      // D
        }

        if (q == 0) {
            outs[((size_t)b * NTOK + n) * DD + d] = sState[d];
        }
        // next token's sTok write is ordered by the barrier at loop top
    }
}

// ---------------------------------------------------------------------------
// Kernel 2: output projection [8192,128] x [128,32000] + bias, f32 WMMA.
// Block = 256 threads = 8 waves arranged 2 (M) x 4 (N); each wave owns a
// 32x32 patch = 2x2 tiles of 16x16 -> 4 independent accumulator chains,
// each fragment reused twice (halves L2 read amplification to 8:1).
// K=128 -> 32 k-steps of V_WMMA_F32_16X16X4_F32 per accumulator.
// A frag (16x4 f32): lane l -> row m=l&15, k = 2*(l>>4)+c (c = vgpr idx).
// B frag (4x16 f32): lane l -> col n=l&15, k = 2*(l>>4)+c.
// D (16x16 f32): vgpr r, lane l -> row r+8*(l>>4), col l&15.
// ---------------------------------------------------------------------------
__launch_bounds__(256)
__global__ void proj_kernel(const float* __restrict__ outs,   // [8192,128]
                            const float* __restrict__ out_w,  // [32000,128]
                            const float* __restrict__ out_b,  // [32000]
                            float*       __restrict__ out)    // [8192,32000]
{
    const int lane = threadIdx.x & 31;
    const int wave = threadIdx.x >> 5;        // 0..7
    const int wy   = wave >> 2;               // 0..1  (M direction)
    const int wx   = wave & 3;                // 0..3  (N direction)
    const int half = lane >> 4;               // 0/1
    const int mr   = lane & 15;
    const int m0   = blockIdx.y * 64 + wy * 32;
    const int v0   = blockIdx.x * 128 + wx * 32;

    const float* a0 = outs  + (size_t)(m0 + mr) * DD + 2 * half;
    const float* a1 = a0 + (size_t)16 * DD;
    const float* b0 = out_w + (size_t)(v0 + mr) * DD + 2 * half;
    const float* b1 = b0 + (size_t)16 * DD;

    v8f acc00 = {0.f,0.f,0.f,0.f,0.f,0.f,0.f,0.f};
    v8f acc01 = acc00, acc10 = acc00, acc11 = acc00;

    #pragma unroll
    for (int kk = 0; kk < 32; ++kk) {
        v2f A0 = *(const v2f*)(a0 + kk * 4);
        v2f A1 = *(const v2f*)(a1 + kk * 4);
        v2f B0 = *(const v2f*)(b0 + kk * 4);
        v2f B1 = *(const v2f*)(b1 + kk * 4);
        acc00 = __builtin_amdgcn_wmma_f32_16x16x4_f32(false, A0, false, B0,
                                                      (short)0, acc00, false, false);
        acc01 = __builtin_amdgcn_wmma_f32_16x16x4_f32(false, A0, false, B1,
                                                      (short)0, acc01, false, false);
        acc10 = __builtin_amdgcn_wmma_f32_16x16x4_f32(false, A1, false, B0,
                                                      (short)0, acc10, false, false);
        acc11 = __builtin_amdgcn_wmma_f32_16x16x4_f32(false, A1, false, B1,
                                                      (short)0, acc11, false, false);
    }

    const float bias0 = out_b[v0 + mr];
    const float bias1 = out_b[v0 + 16 + mr];

    #pragma unroll
    for (int r = 0; r < 8; ++r) {
        const int ma = m0 + r + 8 * half;
        const int mb = ma + 16;
        out[(size_t)ma * VV + (v0 + mr)]      = acc00[r] + bias0;
        out[(size_t)ma * VV + (v0 + 16 + mr)] = acc01[r] + bias1;
        out[(size_t)mb * VV + (v0 + mr)]      = acc10[r] + bias0;
        out[(size_t)mb * VV + (v0 + 16 + mr)] = acc11[r] + bias1;
    }
}

// ---------------------------------------------------------------------------
extern "C" void kernel_launch(void* const* d_in, const int* in_sizes, int n_in,
                              void* d_out, int out_size, void* d_ws, size_t ws_size,
                              hipStream_t stream)
{
    const int*   input_ids = (const int*)  d_in[0];
    const float* embed     = (const float*)d_in[1];
    const float* W         = (const float*)d_in[2];
    const float* gate_w    = (const float*)d_in[3];
    const float* gate_b    = (const float*)d_in[4];
    const float* out_w     = (const float*)d_in[5];
    const float* out_b     = (const float*)d_in[6];
    const int*   n_inner   = (const int*)  d_in[7];

    float* outbuf = (float*)d_out;
    float* outs   = (float*)d_ws;     // [4*2048*128] f32 = 4 MB

    rnn_kernel<<<dim3(BSEQ), dim3(512), 0, stream>>>(
        input_ids, embed, W, gate_w, gate_b, n_inner, outs);
    proj_kernel<<<dim3(VV / 128, (BSEQ * NTOK) / 64), dim3(256), 0, stream>>>(
        outs, out_w, out_b, outbuf);
}